// Improved_DSTGAT_Attn_Adaptive_84061099918017
// MI455X (gfx1250) — compile-verified
//
#include <hip/hip_runtime.h>
#include <math.h>

typedef float v2f __attribute__((ext_vector_type(2)));
typedef float v8f __attribute__((ext_vector_type(8)));

#define WMMA_F32_4(a, b, c) \
    __builtin_amdgcn_wmma_f32_16x16x4_f32(false, (a), false, (b), (short)0, (c), false, false)

static __device__ __forceinline__ v8f vzero8() {
    v8f z;
    #pragma unroll
    for (int i = 0; i < 8; ++i) z[i] = 0.0f;
    return z;
}

// branchless ELU on hardware v_exp_f32 (no exec-mask branching, no libm expansion)
static __device__ __forceinline__ float elu1(float x) {
    float e = __expf(x) - 1.0f;
    return x > 0.0f ? x : e;
}

// ---------------------------------------------------------------------------
// Kernel 1: fused per-(b,c) feature extractor.
// One block of 64 threads (2 waves) per signal of length 500.
// x row staged into LDS with GLOBAL_LOAD_ASYNC_TO_LDS_B128 (ASYNCcnt path);
// conv1 (K=25, pad 12) and the 16x16 pointwise conv run on V_WMMA_F32_16X16X4_F32.
// ---------------------------------------------------------------------------
#define TLEN 500
#define TPAD 512
#define XOFF 12

__global__ void feat_kernel(const float* __restrict__ x,
                            const float* __restrict__ conv1_w,   // [16*25]
                            const float* __restrict__ bn1,       // [4*16]
                            const float* __restrict__ dw_w,      // [16*3]
                            const float* __restrict__ bn2,       // [4*16]
                            const float* __restrict__ sdw_w,     // [16*3]
                            const float* __restrict__ spw_w,     // [16*16]
                            const float* __restrict__ bn3,       // [4*16]
                            float* __restrict__ feats)           // [8192*16]
{
    __shared__ __align__(16) float xpad[XOFF + TPAD + 32];  // xpad[t+12] = x[t]
    __shared__ float Abuf[16 * TPAD];
    __shared__ float Bbuf[16 * TPAD];
    __shared__ float sc1[16], sh1[16], sc2[16], sh2[16], sc3[16], sh3[16];
    __shared__ float featsum[16];

    const int bc   = blockIdx.x;
    const int tid  = threadIdx.x;          // 0..63
    const int lane = tid & 31;
    const int wv   = tid >> 5;             // wave id 0/1
    const int half = lane >> 4;            // 0: K pair {0,1}; 1: K pair {2,3}
    const int lrow = lane & 15;

    const float* xr = x + (long)bc * TLEN;

    // ---- async DMA: 125 x B128 = 500 floats -> xpad[XOFF..XOFF+500) ----
    {
        unsigned ldsbase = (unsigned)(unsigned long long)(uintptr_t)&xpad[0];
        for (int i = tid; i < 125; i += 64) {
            unsigned lds = ldsbase + (unsigned)((XOFF + i * 4) * 4);   // 16B aligned
            unsigned long long ga = (unsigned long long)(uintptr_t)(xr + i * 4);
            asm volatile("global_load_async_to_lds_b128 %0, %1, off"
                         :: "v"(lds), "v"(ga) : "memory");
        }
    }
    // zero-pad head/tail with plain DS stores (disjoint from the DMA region)
    for (int i = tid; i < XOFF; i += 64) xpad[i] = 0.0f;
    for (int i = tid; i < (TPAD + 32 - TLEN); i += 64) xpad[XOFF + TLEN + i] = 0.0f;

    if (tid < 16) {
        float g, b, m, v, s;
        g = bn1[tid]; b = bn1[16 + tid]; m = bn1[32 + tid]; v = bn1[48 + tid];
        s = g * rsqrtf(v + 1e-5f); sc1[tid] = s; sh1[tid] = b - m * s;
        g = bn2[tid]; b = bn2[16 + tid]; m = bn2[32 + tid]; v = bn2[48 + tid];
        s = g * rsqrtf(v + 1e-5f); sc2[tid] = s; sh2[tid] = b - m * s;
        g = bn3[tid]; b = bn3[16 + tid]; m = bn3[32 + tid]; v = bn3[48 + tid];
        s = g * rsqrtf(v + 1e-5f); sc3[tid] = s; sh3[tid] = b - m * s;
        featsum[tid] = 0.0f;
    }
    asm volatile("s_wait_asynccnt 0x0" ::: "memory");   // DMA -> LDS visible
    __syncthreads();

    // ---- conv1 as WMMA: out[ch,t] = sum_k w[ch,k] * xpad[t+k] ; K padded 25->28 ----
    v2f afrag[7];
    #pragma unroll
    for (int kk = 0; kk < 7; ++kk) {
        int k0 = kk * 4 + half * 2;
        afrag[kk].x = (k0     < 25) ? conv1_w[lrow * 25 + k0]     : 0.0f;
        afrag[kk].y = (k0 + 1 < 25) ? conv1_w[lrow * 25 + k0 + 1] : 0.0f;
    }
    for (int nt = wv; nt < 32; nt += 2) {      // uniform per wave
        const int t = nt * 16 + lrow;
        v8f c = vzero8();
        #pragma unroll
        for (int kk = 0; kk < 7; ++kk) {
            int k0 = kk * 4 + half * 2;
            v2f bfr;
            bfr.x = xpad[t + k0];
            bfr.y = xpad[t + k0 + 1];
            c = WMMA_F32_4(afrag[kk], bfr, c);
        }
        #pragma unroll
        for (int v = 0; v < 8; ++v) {
            int ch = v + half * 8;
            float y = elu1(c[v] * sc1[ch] + sh1[ch]);
            Abuf[ch * TPAD + t] = (t < TLEN) ? y : 0.0f;   // zero tail for dw conv
        }
    }
    __syncthreads();

    // ---- depthwise conv K=3 pad=1 + BN + ELU (VALU) ----
    for (int idx = tid; idx < 16 * TLEN; idx += 64) {
        int ch = idx / TLEN;
        int t  = idx - ch * TLEN;
        const float* w = dw_w + ch * 3;
        float acc = w[1] * Abuf[ch * TPAD + t];
        if (t > 0)        acc += w[0] * Abuf[ch * TPAD + t - 1];
        if (t < TLEN - 1) acc += w[2] * Abuf[ch * TPAD + t + 1];
        Bbuf[ch * TPAD + t] = elu1(acc * sc2[ch] + sh2[ch]);
    }
    for (int idx = tid; idx < 16 * (TPAD - TLEN); idx += 64) {
        int ch = idx / (TPAD - TLEN);
        int t  = TLEN + (idx - ch * (TPAD - TLEN));
        Bbuf[ch * TPAD + t] = 0.0f;
    }
    __syncthreads();

    // ---- separable depthwise K=3 pad=1 (no act) -> Abuf ----
    for (int idx = tid; idx < 16 * TLEN; idx += 64) {
        int ch = idx / TLEN;
        int t  = idx - ch * TLEN;
        const float* w = sdw_w + ch * 3;
        float acc = w[1] * Bbuf[ch * TPAD + t];
        if (t > 0)        acc += w[0] * Bbuf[ch * TPAD + t - 1];
        if (t < TLEN - 1) acc += w[2] * Bbuf[ch * TPAD + t + 1];
        Abuf[ch * TPAD + t] = acc;
    }
    for (int idx = tid; idx < 16 * (TPAD - TLEN); idx += 64) {
        int ch = idx / (TPAD - TLEN);
        int t  = TLEN + (idx - ch * (TPAD - TLEN));
        Abuf[ch * TPAD + t] = 0.0f;
    }
    __syncthreads();

    // ---- pointwise 16x16 conv via WMMA + BN + ELU + time-mean ----
    v2f sfrag[4];
    #pragma unroll
    for (int kk = 0; kk < 4; ++kk) {
        int k0 = kk * 4 + half * 2;
        sfrag[kk].x = spw_w[lrow * 16 + k0];
        sfrag[kk].y = spw_w[lrow * 16 + k0 + 1];
    }
    v8f acc8 = vzero8();
    for (int nt = wv; nt < 32; nt += 2) {
        const int t = nt * 16 + lrow;
        v8f c = vzero8();
        #pragma unroll
        for (int kk = 0; kk < 4; ++kk) {
            int k0 = kk * 4 + half * 2;
            v2f bfr;
            bfr.x = Abuf[k0 * TPAD + t];
            bfr.y = Abuf[(k0 + 1) * TPAD + t];
            c = WMMA_F32_4(sfrag[kk], bfr, c);
        }
        if (t < TLEN) {
            #pragma unroll
            for (int v = 0; v < 8; ++v) {
                int ch = v + half * 8;
                acc8[v] += elu1(c[v] * sc3[ch] + sh3[ch]);
            }
        }
    }
    #pragma unroll
    for (int v = 0; v < 8; ++v) {
        int ch = v + half * 8;
        atomicAdd(&featsum[ch], acc8[v]);
    }
    __syncthreads();
    if (tid < 16) feats[bc * 16 + tid] = featsum[tid] * (1.0f / 500.0f);
}

// ---------------------------------------------------------------------------
// Kernel 2: graph construction -> logM[i][j]  (64x64, -1e30 where no edge)
// ---------------------------------------------------------------------------
__global__ void graph_kernel(const float* __restrict__ emb,      // [64*16]
                             const int* __restrict__ edges,      // [2*256]
                             float* __restrict__ logM)           // [64*64]
{
    __shared__ float E[64 * 16];
    __shared__ float Mcnt[64 * 64];
    const int tid = threadIdx.x;   // 64 threads
    for (int i = tid; i < 64 * 16; i += 64) E[i] = emb[i];
    for (int i = tid; i < 64 * 64; i += 64) Mcnt[i] = 0.0f;
    __syncthreads();
    for (int e = tid; e < 256; e += 64) {
        int s = edges[e];
        int d = edges[256 + e];
        atomicAdd(&Mcnt[s * 64 + d], 1.0f);
    }
    __syncthreads();
    const int i = tid;
    float arow[64];
    float rs = 0.0f;
    for (int j = 0; j < 64; ++j) {
        float dot = 0.0f;
        for (int k = 0; k < 16; ++k) dot += E[i * 16 + k] * E[j * 16 + k];
        float a = dot > 0.0f ? dot : 0.0f;
        arow[j] = a;
        rs += a;
    }
    float inv = 1.0f / (rs + 1e-6f);
    for (int j = 0; j < 64; ++j) {
        float dyn = (arow[j] * inv > 0.1f) ? 1.0f : 0.0f;
        float M = (i == j) ? 1.0f : (Mcnt[i * 64 + j] + dyn);
        logM[i * 64 + j] = (M > 0.0f) ? __logf(M) : -1e30f;
    }
}

// ---------------------------------------------------------------------------
// Kernel 3/5: fused GAT layer (one wave per batch element).
// xl matmul and per-head aggregation on V_WMMA_F32_16X16X4_F32.
// ---------------------------------------------------------------------------
template <int IN>
__global__ void gat_kernel(const float* __restrict__ xin_g,   // [128*64*IN]
                           const float* __restrict__ gate,    // [IN] or nullptr
                           const float* __restrict__ Wg,      // [IN*128]
                           const float* __restrict__ asrc,    // [4*32]
                           const float* __restrict__ adst,    // [4*32]
                           const float* __restrict__ bias,    // [32]
                           const float* __restrict__ bnp,     // [4*32]
                           const float* __restrict__ skip_w,  // [32*IN] or nullptr
                           const float* __restrict__ logM,    // [64*64]
                           float* __restrict__ hout)          // [128*64*32]
{
    __shared__ float xin_s[64 * IN];
    __shared__ float W_s[IN * 128];
    __shared__ float xl_s[64 * 128];
    __shared__ float s_s[64 * 4], d_s[64 * 4];
    __shared__ float alpha_s[64 * 64];
    __shared__ float bnsc[32], bnsh[32];

    const int b    = blockIdx.x;
    const int lane = threadIdx.x;      // 32 threads = 1 wave
    const int half = lane >> 4;
    const int lrow = lane & 15;

    for (int i = lane; i < 64 * IN; i += 32) {
        float g = gate ? gate[i % IN] : 1.0f;
        xin_s[i] = xin_g[(long)b * 64 * IN + i] * g;
    }
    for (int i = lane; i < IN * 128; i += 32) W_s[i] = Wg[i];
    {
        float g = bnp[lane], bb = bnp[32 + lane], m = bnp[64 + lane], v = bnp[96 + lane];
        float s = g * rsqrtf(v + 1e-5f);
        bnsc[lane] = s;
        bnsh[lane] = bb - m * s;
    }
    __syncthreads();

    // ---- xl = xin @ W : [64 x 128] via WMMA ----
    for (int mt = 0; mt < 4; ++mt) {
        for (int nt = 0; nt < 8; ++nt) {
            v8f c = vzero8();
            #pragma unroll
            for (int kk = 0; kk < IN / 4; ++kk) {
                int k0 = kk * 4 + half * 2;
                v2f a, bfr;
                a.x = xin_s[(mt * 16 + lrow) * IN + k0];
                a.y = xin_s[(mt * 16 + lrow) * IN + k0 + 1];
                bfr.x = W_s[k0 * 128 + nt * 16 + lrow];
                bfr.y = W_s[(k0 + 1) * 128 + nt * 16 + lrow];
                c = WMMA_F32_4(a, bfr, c);
            }
            #pragma unroll
            for (int v = 0; v < 8; ++v)
                xl_s[(mt * 16 + v + half * 8) * 128 + nt * 16 + lrow] = c[v];
        }
    }
    __syncthreads();

    // ---- attention source/dest terms ----
    for (int p = lane; p < 256; p += 32) {
        int j = p >> 2, h = p & 3;
        float ss = 0.0f, dd = 0.0f;
        for (int f = 0; f < 32; ++f) {
            float xv = xl_s[j * 128 + h * 32 + f];
            ss += xv * asrc[h * 32 + f];
            dd += xv * adst[h * 32 + f];
        }
        s_s[p] = ss;
        d_s[p] = dd;
    }
    __syncthreads();

    // ---- per-head softmax + aggregation, accumulated across heads ----
    v8f acc[4][2];
    #pragma unroll
    for (int mt = 0; mt < 4; ++mt)
        #pragma unroll
        for (int nt = 0; nt < 2; ++nt) acc[mt][nt] = vzero8();

    for (int h = 0; h < 4; ++h) {
        for (int rr = 0; rr < 2; ++rr) {
            int i = lane + rr * 32;
            float di = d_s[i * 4 + h];
            const float* lm = logM + i * 64;
            float mx = -1e30f;
            for (int j = 0; j < 64; ++j) {
                float e = di + s_s[j * 4 + h];
                e = (e > 0.0f) ? e : 0.2f * e;
                mx = fmaxf(mx, e + lm[j]);
            }
            float sum = 0.0f;
            for (int j = 0; j < 64; ++j) {
                float e = di + s_s[j * 4 + h];
                e = (e > 0.0f) ? e : 0.2f * e;
                float ex = __expf(e + lm[j] - mx);
                alpha_s[i * 64 + j] = ex;
                sum += ex;
            }
            float inv = 1.0f / sum;
            for (int j = 0; j < 64; ++j) alpha_s[i * 64 + j] *= inv;
        }
        __syncthreads();
        for (int mt = 0; mt < 4; ++mt) {
            for (int nt = 0; nt < 2; ++nt) {
                #pragma unroll
                for (int kk = 0; kk < 16; ++kk) {
                    int k0 = kk * 4 + half * 2;
                    v2f a, bfr;
                    a.x = alpha_s[(mt * 16 + lrow) * 64 + k0];
                    a.y = alpha_s[(mt * 16 + lrow) * 64 + k0 + 1];
                    bfr.x = xl_s[k0 * 128 + h * 32 + nt * 16 + lrow];
                    bfr.y = xl_s[(k0 + 1) * 128 + h * 32 + nt * 16 + lrow];
                    acc[mt][nt] = WMMA_F32_4(a, bfr, acc[mt][nt]);
                }
            }
        }
        __syncthreads();
    }

    // ---- epilogue: head-mean + bias + BN + skip + GELU ----
    for (int mt = 0; mt < 4; ++mt) {
        for (int nt = 0; nt < 2; ++nt) {
            #pragma unroll
            for (int v = 0; v < 8; ++v) {
                int i = mt * 16 + v + half * 8;
                int f = nt * 16 + lrow;
                float g = acc[mt][nt][v] * 0.25f + bias[f];
                g = g * bnsc[f] + bnsh[f];
                float sk;
                if (skip_w) {
                    sk = 0.0f;
                    for (int k = 0; k < IN; ++k) sk += xin_s[i * IN + k] * skip_w[f * IN + k];
                } else {
                    sk = xin_s[i * IN + f];
                }
                float z = g + sk;
                float out = 0.5f * z * (1.0f + erff(z * 0.70710678f));
                hout[((long)b * 64 + i) * 32 + f] = out;
            }
        }
    }
}

// ---------------------------------------------------------------------------
// Kernel 4/6a: SE gate (global mean over 8192 rows -> squeeze -> sigmoid)
// ---------------------------------------------------------------------------
__global__ void se_kernel(const float* __restrict__ h,    // [8192*32]
                          const float* __restrict__ w1,   // [8*32]
                          const float* __restrict__ w2,   // [32*8]
                          float* __restrict__ gate)       // [32]
{
    __shared__ float part[256];
    __shared__ float m[32];
    __shared__ float q[8];
    const int tid = threadIdx.x;       // 256 threads
    const int f = tid & 31, grp = tid >> 5;
    float sum = 0.0f;
    for (int r = grp; r < 8192; r += 8) sum += h[(long)r * 32 + f];
    part[tid] = sum;
    __syncthreads();
    if (tid < 32) {
        float s = 0.0f;
        for (int g = 0; g < 8; ++g) s += part[g * 32 + tid];
        m[tid] = s * (1.0f / 8192.0f);
    }
    __syncthreads();
    if (tid < 8) {
        float a = 0.0f;
        for (int k = 0; k < 32; ++k) a += m[k] * w1[tid * 32 + k];
        q[tid] = a > 0.0f ? a : 0.0f;
    }
    __syncthreads();
    if (tid < 32) {
        float a = 0.0f;
        for (int k = 0; k < 8; ++k) a += q[k] * w2[tid * 8 + k];
        gate[tid] = 1.0f / (1.0f + __expf(-a));
    }
}

// ---------------------------------------------------------------------------
// Kernel 6b: SE2 scale + mean pool over nodes + classifier
// ---------------------------------------------------------------------------
__global__ void final_kernel(const float* __restrict__ h2,     // [128*64*32]
                             const float* __restrict__ gate2,  // [32]
                             const float* __restrict__ clf_w,  // [2*32]
                             const float* __restrict__ clf_b,  // [2]
                             float* __restrict__ out)          // [128*2]
{
    __shared__ float pooled[32];
    const int b = blockIdx.x, f = threadIdx.x;   // 32 threads
    float s = 0.0f;
    for (int i = 0; i < 64; ++i) s += h2[((long)b * 64 + i) * 32 + f];
    pooled[f] = (s * (1.0f / 64.0f)) * gate2[f];
    __syncthreads();
    if (f < 2) {
        float a = clf_b[f];
        for (int k = 0; k < 32; ++k) a += pooled[k] * clf_w[f * 32 + k];
        out[b * 2 + f] = a;
    }
}

// ---------------------------------------------------------------------------
extern "C" void kernel_launch(void* const* d_in, const int* in_sizes, int n_in,
                              void* d_out, int out_size, void* d_ws, size_t ws_size,
                              hipStream_t stream) {
    (void)in_sizes; (void)n_in; (void)out_size; (void)ws_size;
    const float* x          = (const float*)d_in[0];
    const int*   edges      = (const int*)  d_in[1];
    const float* conv1_w    = (const float*)d_in[2];
    const float* bn_c1      = (const float*)d_in[3];
    const float* dw_w       = (const float*)d_in[4];
    const float* bn_c2      = (const float*)d_in[5];
    const float* sdw_w      = (const float*)d_in[6];
    const float* spw_w      = (const float*)d_in[7];
    const float* bn_c3      = (const float*)d_in[8];
    const float* node_embed = (const float*)d_in[9];
    const float* gat1_w     = (const float*)d_in[10];
    const float* gat1_asrc  = (const float*)d_in[11];
    const float* gat1_adst  = (const float*)d_in[12];
    const float* gat1_bias  = (const float*)d_in[13];
    const float* bn_g1      = (const float*)d_in[14];
    const float* skip1_w    = (const float*)d_in[15];
    const float* se1_w1     = (const float*)d_in[16];
    const float* se1_w2     = (const float*)d_in[17];
    const float* gat2_w     = (const float*)d_in[18];
    const float* gat2_asrc  = (const float*)d_in[19];
    const float* gat2_adst  = (const float*)d_in[20];
    const float* gat2_bias  = (const float*)d_in[21];
    const float* bn_g2      = (const float*)d_in[22];
    const float* se2_w1     = (const float*)d_in[23];
    const float* se2_w2     = (const float*)d_in[24];
    const float* clf_w      = (const float*)d_in[25];
    const float* clf_b      = (const float*)d_in[26];
    float* out = (float*)d_out;

    float* ws    = (float*)d_ws;
    float* feats = ws;                  // 8192*16
    float* logM  = feats + 8192 * 16;   // 64*64
    float* h1    = logM + 64 * 64;      // 8192*32
    float* gate1 = h1 + 8192 * 32;      // 32 (+ pad)
    float* h2    = gate1 + 64;          // 8192*32
    float* gate2 = h2 + 8192 * 32;      // 32

    feat_kernel<<<128 * 64, 64, 0, stream>>>(x, conv1_w, bn_c1, dw_w, bn_c2,
                                             sdw_w, spw_w, bn_c3, feats);
    graph_kernel<<<1, 64, 0, stream>>>(node_embed, edges, logM);
    gat_kernel<16><<<128, 32, 0, stream>>>(feats, nullptr, gat1_w, gat1_asrc,
                                           gat1_adst, gat1_bias, bn_g1, skip1_w,
                                           logM, h1);
    se_kernel<<<1, 256, 0, stream>>>(h1, se1_w1, se1_w2, gate1);
    gat_kernel<32><<<128, 32, 0, stream>>>(h1, gate1, gat2_w, gat2_asrc,
                                           gat2_adst, gat2_bias, bn_g2, nullptr,
                                           logM, h2);
    se_kernel<<<1, 256, 0, stream>>>(h2, se2_w1, se2_w2, gate2);
    final_kernel<<<128, 32, 0, stream>>>(h2, gate2, clf_w, clf_b, out);
}